// CSCFCLayer_19559281066793
// MI455X (gfx1250) — compile-verified
//
#include <hip/hip_runtime.h>
#include <hip/hip_bf16.h>

typedef __attribute__((ext_vector_type(2))) float v2f;
typedef __attribute__((ext_vector_type(8))) float v8f;

#define C_DIM 8192
#define N_DIM 8192
#define F_DIM 64
#define B_DIM 128
#define TILE_N 16
#define TILE_B 16
#define KDIM 80            // TILE_N-1 + F_DIM = 79, padded to 80 (multiple of 4)
#define LDS_STRIDE 84      // 80 + 4 pad: 84 mod 64 = 20, gcd(20,64)=4 -> conflict-free

__global__ __launch_bounds__(256)
void cscfc_wmma_kernel(const float* __restrict__ x,
                       const float* __restrict__ kern,
                       const float* __restrict__ bias,
                       float* __restrict__ out) {
    // Both tiles use the same padded K-major stride so every WMMA fragment is
    // one aligned ds_load_b64 (two adjacent K values per lane).
    __shared__ float xs [B_DIM  * LDS_STRIDE];  // xs[b][k]  = x[b][(n0+k)%C]
    __shared__ float w2t[TILE_N * LDS_STRIDE];  // w2t[j][k] = banded kernel tile, transposed

    const int tid = threadIdx.x;
    const int n0  = blockIdx.x * TILE_N;

    // CDNA5 speculative prefetch of this tile's first weight rows (global_prefetch_b8)
    __builtin_prefetch(&kern[(size_t)((n0 + (tid & 63)) & (C_DIM - 1)) * N_DIM + n0], 0, 0);

    // Stage banded weight tile (transposed): w2t[j][k] = kernel[(n0+k)%C][n0+j]
    // masked to 0 <= k-j < 64. Global side: 16 contiguous floats per row -> coalesced.
    for (int i = tid; i < KDIM * TILE_N; i += 256) {
        int k   = i / TILE_N;
        int j   = i % TILE_N;
        int row = (n0 + k) & (C_DIM - 1);
        float v = kern[(size_t)row * N_DIM + (size_t)(n0 + j)];
        int f   = k - j;
        w2t[j * LDS_STRIDE + k] = (f >= 0 && f < F_DIM) ? v : 0.0f;
    }

    // Stage x slice: xs[b][k] = x[b][(n0+k)%C]; 80 contiguous floats per row (mod wrap).
    for (int i = tid; i < B_DIM * KDIM; i += 256) {
        int b   = i / KDIM;
        int k   = i % KDIM;
        int col = (n0 + k) & (C_DIM - 1);
        xs[b * LDS_STRIDE + k] = x[(size_t)b * C_DIM + (size_t)col];
    }

    __syncthreads();

    // Each of the 8 waves computes one 16(b) x 16(n) tile: D = A(16x80) * B(80x16)
    const int wave = tid >> 5;
    const int lane = tid & 31;
    const int half = lane >> 4;   // K phase select per ISA 16x4 f32 layout
    const int lj   = lane & 15;   // M for A-frag, N for B/C/D-frags
    const int b0   = wave * TILE_B;

    const float* arow = &xs [(b0 + lj) * LDS_STRIDE];  // A: lane owns row M=lj
    const float* brow = &w2t[lj        * LDS_STRIDE];  // B: lane owns col N=lj

    v8f acc = {};
#pragma unroll
    for (int kk = 0; kk < KDIM / 4; ++kk) {
        const int k0 = kk * 4 + 2 * half;  // VGPR pair holds K={k0, k0+1}
        // (row*84 + k0) is even -> 8-byte aligned ds_load_b64 into an even VGPR pair
        v2f a = *(const v2f*)(arow + k0);
        v2f b = *(const v2f*)(brow + k0);
        acc = __builtin_amdgcn_wmma_f32_16x16x4_f32(
            /*neg_a=*/false, a, /*neg_b=*/false, b,
            /*c_mod=*/(short)0, acc, /*reuse_a=*/false, /*reuse_b=*/false);
    }

    // C/D 16x16 f32 layout: VGPR r holds M = r + 8*half, N = lj
    const float bv = bias[n0 + lj];
#pragma unroll
    for (int r = 0; r < 8; ++r) {
        int m = r + (half ? 8 : 0);
        out[(size_t)(b0 + m) * N_DIM + (size_t)(n0 + lj)] = acc[r] + bv;
    }
}

extern "C" void kernel_launch(void* const* d_in, const int* in_sizes, int n_in,
                              void* d_out, int out_size, void* d_ws, size_t ws_size,
                              hipStream_t stream) {
    (void)in_sizes; (void)n_in; (void)out_size; (void)d_ws; (void)ws_size;
    const float* x    = (const float*)d_in[0];   // (B, C) f32
    const float* kern = (const float*)d_in[1];   // (C, N) f32
    const float* bias = (const float*)d_in[2];   // (N,)  f32
    float* out        = (float*)d_out;           // (B, N) f32

    dim3 grid(N_DIM / TILE_N);   // 512 blocks, one 16-wide n-tile each
    dim3 block(256);             // 8 wave32s = 8 b-tiles covering B=128
    hipLaunchKernelGGL(cscfc_wmma_kernel, grid, block, 0, stream, x, kern, bias, out);
}